// Set2Set_83056077570515
// MI455X (gfx1250) — compile-verified
//
#include <hip/hip_runtime.h>
#include <hip/hip_bf16.h>

typedef float v2f __attribute__((ext_vector_type(2)));
typedef float v8f __attribute__((ext_vector_type(8)));

#define HIDDEN 256
#define NGRAPH 1024
#define N_ITERS 6
#define RCHUNK 512

__device__ __forceinline__ float sigmoidf_(float x) { return 1.0f / (1.0f + __expf(-x)); }

// ---------------------------------------------------------------------------
// WMMA fp32 GEMM:  C[M,N] = A[M,K] @ B[K,N] + bias[N], with B given K-major
// (Bt[k*ldbt + n]). One wave per 16x64 output tile; V_WMMA_F32_16X16X4_F32.
// A fragment (16x4):  lane -> M=lane&15, K=kb..kb+1, kb=(lane>>4)*2
// B fragment (4x16):  lane -> N=lane&15, K=kb..kb+1   (coalesced: consecutive
//                     lanes read consecutive columns of Bt)
// D (16x16, 8 VGPRs): vgpr v -> M = v + (lane>=16 ? 8 : 0), N = lane&15
// Loads are batched per K-chunk of 16 (4 A frags + 16 B frags) so the
// compiler can overlap L2 latency with the 16-WMMA chain instead of
// emitting s_wait_loadcnt 0 per fragment.
// ---------------------------------------------------------------------------
__global__ __launch_bounds__(32)
void k_wmma_gemm(const float* __restrict__ A, int lda,
                 const float* __restrict__ Bt, int ldbt,
                 const float* __restrict__ bias,
                 float* __restrict__ C, int ldc, int K) {
  const int tn   = blockIdx.x * 64;
  const int tm   = blockIdx.y * 16;
  const int lane = threadIdx.x;
  const int r    = lane & 15;
  const int kb   = (lane >> 4) << 1;

  v8f acc[4] = {};
  const float* ap0 = A + (size_t)(tm + r) * lda + kb;

  for (int k0 = 0; k0 < K; k0 += 16) {
    v2f a[4];
    v2f b[4][4];
#pragma unroll
    for (int kk = 0; kk < 4; ++kk) {
      const float* ap = ap0 + k0 + kk * 4;
      a[kk].x = ap[0];
      a[kk].y = ap[1];
      const float* bp = Bt + (size_t)(k0 + kk * 4 + kb) * ldbt + tn + r;
#pragma unroll
      for (int n = 0; n < 4; ++n) {
        b[kk][n].x = bp[n * 16];          // row k0+kk*4+kb
        b[kk][n].y = bp[n * 16 + ldbt];   // row k0+kk*4+kb+1
      }
    }
#pragma unroll
    for (int kk = 0; kk < 4; ++kk) {
#pragma unroll
      for (int n = 0; n < 4; ++n) {
        acc[n] = __builtin_amdgcn_wmma_f32_16x16x4_f32(
            /*neg_a=*/false, a[kk], /*neg_b=*/false, b[kk][n],
            /*c_mod=*/(short)0, acc[n], /*reuse_a=*/false, /*reuse_b=*/false);
      }
    }
  }

  const int rowoff = (lane >> 4) * 8;
#pragma unroll
  for (int n = 0; n < 4; ++n) {
    const int col = tn + n * 16 + r;
    const float bv = bias ? bias[col] : 0.0f;
#pragma unroll
    for (int v = 0; v < 8; ++v) {
      const int row = tm + rowoff + v;
      C[(size_t)row * ldc + col] = acc[n][v] + bv;
    }
  }
}

// ---------------------------------------------------------------------------
// Prep (one-time per launch):
//   WcatT[768,1024]: K-major concat of W_ih|W_hh  (WcatT[k,j] = Wcat[j,k])
//   fc_wT[512,256] : K-major fc weights           (fc_wT[k,j] = fc_w[j,k])
//   biascat = b_ih + b_hh ; zero Acat / h / c.
// ---------------------------------------------------------------------------
__global__ void k_prep(const float* __restrict__ W_ih, const float* __restrict__ W_hh,
                       const float* __restrict__ b_ih, const float* __restrict__ b_hh,
                       const float* __restrict__ fc_w,
                       float* __restrict__ WcatT, float* __restrict__ fc_wT,
                       float* __restrict__ biascat,
                       float* __restrict__ Acat, float* __restrict__ h,
                       float* __restrict__ c) {
  const int total = 3 * HIDDEN * 4 * HIDDEN;  // 768*1024
  for (int idx = blockIdx.x * blockDim.x + threadIdx.x; idx < total;
       idx += gridDim.x * blockDim.x) {
    const int k = idx >> 10;           // 0..767
    const int j = idx & (4 * HIDDEN - 1);  // 0..1023
    WcatT[idx] = (k < 2 * HIDDEN) ? W_ih[(size_t)j * 2 * HIDDEN + k]
                                  : W_hh[(size_t)j * HIDDEN + (k - 2 * HIDDEN)];
    Acat[idx] = 0.0f;
    if (idx < 2 * HIDDEN * HIDDEN) {   // 512*256 fc_wT
      const int kk = idx >> 8;
      const int jj = idx & (HIDDEN - 1);
      fc_wT[idx] = fc_w[(size_t)jj * 2 * HIDDEN + kk];
    }
    if (idx < 4 * HIDDEN) biascat[idx] = b_ih[idx] + b_hh[idx];
    if (idx < NGRAPH * HIDDEN) { h[idx] = 0.0f; c[idx] = 0.0f; }
  }
}

// ---------------------------------------------------------------------------
// LSTM pointwise: gate order i,f,g,o. Writes h, c, and h into Acat
// q-slot (cols 0:256) and h-slot (cols 512:768).
// ---------------------------------------------------------------------------
__global__ void k_lstm_pointwise(const float* __restrict__ gates,
                                 float* __restrict__ h, float* __restrict__ c,
                                 float* __restrict__ Acat) {
  const int total = NGRAPH * HIDDEN;
  for (int idx = blockIdx.x * blockDim.x + threadIdx.x; idx < total;
       idx += gridDim.x * blockDim.x) {
    const int b = idx >> 8;
    const int j = idx & (HIDDEN - 1);
    const float* g = gates + (size_t)b * 4 * HIDDEN;
    const float ig = sigmoidf_(g[j]);
    const float fg = sigmoidf_(g[HIDDEN + j]);
    const float gg = tanhf(g[2 * HIDDEN + j]);
    const float og = sigmoidf_(g[3 * HIDDEN + j]);
    const float cn = fg * c[idx] + ig * gg;
    const float hn = og * tanhf(cn);
    c[idx] = cn;
    h[idx] = hn;
    Acat[(size_t)b * 3 * HIDDEN + j] = hn;               // q part of q_star
    Acat[(size_t)b * 3 * HIDDEN + 2 * HIDDEN + j] = hn;  // h input for next cell
  }
}

// ---------------------------------------------------------------------------
// Per-iteration attention init: readout=0, denom=0, segmax = key(-inf).
// ---------------------------------------------------------------------------
__global__ void k_attn_init(float* __restrict__ readout, float* __restrict__ denom,
                            unsigned* __restrict__ segmax) {
  const int total = NGRAPH * HIDDEN;
  for (int idx = blockIdx.x * blockDim.x + threadIdx.x; idx < total;
       idx += gridDim.x * blockDim.x) {
    readout[idx] = 0.0f;
    if (idx < NGRAPH) {
      denom[idx] = 0.0f;
      segmax[idx] = 0x007FFFFFu;  // monotonic key of -inf
    }
  }
}

// ---------------------------------------------------------------------------
// Pass 1: e[n] = feat[n,:] . h[seg[n],:]; atomicMax segment max (int-keyed).
// One wave per node; 8 floats/lane via two float4 loads (coalesced 1KB/wave).
// ---------------------------------------------------------------------------
__global__ void k_attn_dot(const float* __restrict__ feat, const int* __restrict__ seg,
                           const float* __restrict__ h, float* __restrict__ e,
                           unsigned* __restrict__ segmax, int N) {
  const int lane = threadIdx.x & 31;
  const int wid  = (blockIdx.x * blockDim.x + threadIdx.x) >> 5;
  const int nw   = (gridDim.x * blockDim.x) >> 5;
  for (int n = wid; n < N; n += nw) {
    if (n + nw < N)
      __builtin_prefetch(feat + (size_t)(n + nw) * HIDDEN, 0, 1);
    const int s = seg[n];
    const float4* fp = (const float4*)(feat + (size_t)n * HIDDEN);
    const float4* qp = (const float4*)(h + (size_t)s * HIDDEN);
    const float4 f0 = fp[lane * 2], f1 = fp[lane * 2 + 1];
    const float4 q0 = qp[lane * 2], q1 = qp[lane * 2 + 1];
    float d = f0.x * q0.x + f0.y * q0.y + f0.z * q0.z + f0.w * q0.w +
              f1.x * q1.x + f1.y * q1.y + f1.z * q1.z + f1.w * q1.w;
#pragma unroll
    for (int off = 16; off > 0; off >>= 1) d += __shfl_xor(d, off, 32);
    if (lane == 0) {
      e[n] = d;
      const unsigned bits = __float_as_uint(d);
      const unsigned key = bits ^ ((bits & 0x80000000u) ? 0xFFFFFFFFu : 0x80000000u);
      atomicMax(segmax + s, key);
    }
  }
}

// ---------------------------------------------------------------------------
// Pass 2: ex = exp(e - segmax); denom[seg] += ex  (in-place on e)
// ---------------------------------------------------------------------------
__global__ void k_attn_exp(float* e, const int* __restrict__ seg,
                           const unsigned* __restrict__ segmax,
                           float* __restrict__ denom, int N) {
  const int n = blockIdx.x * blockDim.x + threadIdx.x;
  if (n >= N) return;
  const int s = seg[n];
  const unsigned u = segmax[s];
  const float m = (u & 0x80000000u) ? __uint_as_float(u ^ 0x80000000u)
                                    : __uint_as_float(~u);
  const float v = __expf(e[n] - m);
  e[n] = v;
  atomicAdd(denom + s, v);
}

// Pass 3a: alpha[n] = ex[n] / denom[seg[n]]  (in-place on e)
__global__ void k_attn_alpha(float* e, const int* __restrict__ seg,
                             const float* __restrict__ denom, int N) {
  const int n = blockIdx.x * blockDim.x + threadIdx.x;
  if (n >= N) return;
  e[n] = e[n] / denom[seg[n]];
}

// ---------------------------------------------------------------------------
// Pass 3b: readout[seg] += alpha[n] * feat[n].
// Block of 256 threads owns a contiguous chunk of RCHUNK nodes; thread j owns
// column j; segment_ids are sorted -> register accumulate, atomic flush only
// on segment boundaries.
// ---------------------------------------------------------------------------
__global__ __launch_bounds__(HIDDEN)
void k_attn_readout(const float* __restrict__ feat, const int* __restrict__ seg,
                    const float* __restrict__ alpha, float* __restrict__ readout, int N) {
  const int j = threadIdx.x;
  const int start = blockIdx.x * RCHUNK;
  if (start >= N) return;
  const int end = (start + RCHUNK < N) ? (start + RCHUNK) : N;
  float acc = 0.0f;
  int cur = seg[start];
  for (int n = start; n < end; ++n) {
    const int s = seg[n];
    if (s != cur) {
      atomicAdd(readout + (size_t)cur * HIDDEN + j, acc);
      acc = 0.0f;
      cur = s;
    }
    acc = fmaf(alpha[n], feat[(size_t)n * HIDDEN + j], acc);
  }
  atomicAdd(readout + (size_t)cur * HIDDEN + j, acc);
}

// readout -> Acat cols 256:512 (q_star = [q, readout])
__global__ void k_copy_readout(const float* __restrict__ readout, float* __restrict__ Acat) {
  const int total = NGRAPH * HIDDEN;
  for (int idx = blockIdx.x * blockDim.x + threadIdx.x; idx < total;
       idx += gridDim.x * blockDim.x) {
    const int b = idx >> 8;
    const int j = idx & (HIDDEN - 1);
    Acat[(size_t)b * 3 * HIDDEN + HIDDEN + j] = readout[idx];
  }
}

// ---------------------------------------------------------------------------
extern "C" void kernel_launch(void* const* d_in, const int* in_sizes, int n_in,
                              void* d_out, int out_size, void* d_ws, size_t ws_size,
                              hipStream_t stream) {
  const float* feat = (const float*)d_in[0];
  const int*   seg  = (const int*)d_in[1];
  const float* W_ih = (const float*)d_in[2];
  const float* W_hh = (const float*)d_in[3];
  const float* b_ih = (const float*)d_in[4];
  const float* b_hh = (const float*)d_in[5];
  const float* fc_w = (const float*)d_in[6];
  const float* fc_b = (const float*)d_in[7];
  float* out = (float*)d_out;
  const int N = in_sizes[1];  // number of nodes

  // workspace layout (floats)
  float* ws      = (float*)d_ws;
  float* WcatT   = ws;                              // 768*1024  (K-major)
  float* Acat    = WcatT + 768 * 1024;              // 1024*768  [q | readout | h]
  float* gates   = Acat + 1024 * 768;               // 1024*1024
  float* h       = gates + 1024 * 1024;             // 1024*256
  float* c       = h + NGRAPH * HIDDEN;             // 1024*256
  float* readout = c + NGRAPH * HIDDEN;             // 1024*256
  float* fc_wT   = readout + NGRAPH * HIDDEN;       // 512*256   (K-major)
  float* e       = fc_wT + 2 * HIDDEN * HIDDEN;     // N
  float* denom   = e + N;                           // 1024
  unsigned* segmax = (unsigned*)(denom + NGRAPH);   // 1024
  float* biascat = (float*)(segmax + NGRAPH);       // 1024

  const int nblk_bd = (NGRAPH * HIDDEN + 255) / 256;      // 1024
  const int nblk_n  = (N + 255) / 256;
  const int nblk_rc = (N + RCHUNK - 1) / RCHUNK;

  k_prep<<<3072, 256, 0, stream>>>(W_ih, W_hh, b_ih, b_hh, fc_w,
                                   WcatT, fc_wT, biascat, Acat, h, c);

  const dim3 ggates(4 * HIDDEN / 64, NGRAPH / 16);   // (16, 64)
  const dim3 gfc(HIDDEN / 64, NGRAPH / 16);          // (4, 64)

  for (int it = 0; it < N_ITERS; ++it) {
    // gates = [q_star | h] @ [W_ih | W_hh]^T + (b_ih + b_hh)
    k_wmma_gemm<<<ggates, 32, 0, stream>>>(Acat, 3 * HIDDEN, WcatT, 4 * HIDDEN,
                                           biascat, gates, 4 * HIDDEN, 3 * HIDDEN);
    k_lstm_pointwise<<<nblk_bd, 256, 0, stream>>>(gates, h, c, Acat);
    k_attn_init<<<nblk_bd, 256, 0, stream>>>(readout, denom, segmax);
    k_attn_dot<<<2048, 256, 0, stream>>>(feat, seg, h, e, segmax, N);
    k_attn_exp<<<nblk_n, 256, 0, stream>>>(e, seg, segmax, denom, N);
    k_attn_alpha<<<nblk_n, 256, 0, stream>>>(e, seg, denom, N);
    k_attn_readout<<<nblk_rc, HIDDEN, 0, stream>>>(feat, seg, e, readout, N);
    k_copy_readout<<<nblk_bd, 256, 0, stream>>>(readout, Acat);
  }

  // out = q_star @ fc_w^T + fc_b   (first 512 cols of Acat, row stride 768)
  k_wmma_gemm<<<gfc, 32, 0, stream>>>(Acat, 3 * HIDDEN, fc_wT, HIDDEN,
                                      fc_b, out, HIDDEN, 2 * HIDDEN);
}